// TripletLoss_11321533792629
// MI455X (gfx1250) — compile-verified
//
#include <hip/hip_runtime.h>

// TripletLoss (cosine branch), MI455X / gfx1250, wave32.
//
// Roofline: negatives = 500000 x 128 f32 = 256 MB -> ~11 us at 23.3 TB/s;
// 128 MFLOP of math => pure bandwidth-bound. Stream negatives once; do the
// matvec on the WMMA pipe (V_WMMA_F32_16X16X4_F32, f32-in/f32-acc, exact)
// with the anchor chunk broadcast across all 16 B-columns. Broadcast-B makes
// every D column identical, so a full-wave reduction of
// relu(pos - dot + margin) divided by 16 is layout-robust for C/D.
//
// Grid is sized so each wave handles ~one 16-row tile: no redundant LDS
// B-fragment reloads, one independent 8 KB HBM stream per wave.

#define MARGIN 0.4f

typedef float v8f __attribute__((ext_vector_type(8)));
typedef float v2f __attribute__((ext_vector_type(2)));

// Kernel 1: each wave processes 16-row tiles of `negatives`, accumulating
// sum(relu(pos_sim - dot + margin)) over its tiles; writes one partial per wave.
__global__ __launch_bounds__(256) void triplet_wmma_partial(
    const float* __restrict__ anchor,
    const float* __restrict__ positive,
    const float* __restrict__ negatives,
    float* __restrict__ partials,
    int n_tiles) {
  __shared__ float sA[128];  // anchor staged in LDS (512 B)
  const int tid = threadIdx.x;
  if (tid < 128) sA[tid] = anchor[tid];
  __syncthreads();

  const int lane = tid & 31;
  const int gwave = blockIdx.x * 8 + (tid >> 5);
  const int nwaves = gridDim.x * 8;

  // pos_sim = dot(anchor, positive): 4 elements per lane, butterfly reduce.
  float ps;
  {
    const float* a4 = anchor + lane * 4;
    const float* p4 = positive + lane * 4;
    ps = a4[0] * p4[0] + a4[1] * p4[1] + a4[2] * p4[2] + a4[3] * p4[3];
  }
#pragma unroll
  for (int m = 16; m >= 1; m >>= 1) ps += __shfl_xor(ps, m, 32);

  // A-matrix 16x4 f32 layout: lane L -> row L%16; VGPR0/1 hold K = 2*(L/16)+{0,1}.
  const int half = lane >> 4;  // 0: K={0,1}, 1: K={2,3} within each chunk
  const int mrow = lane & 15;

  float acc = 0.0f;
  for (int t = gwave; t < n_tiles; t += nwaves) {  // normally <=1 iteration
    const float* rowp =
        negatives + (size_t)(t * 16 + mrow) * 128 + (size_t)(half * 2);
    v8f c = {};  // f32 accumulator tile (8 VGPRs)
#pragma unroll 8
    for (int kc = 0; kc < 32; ++kc) {          // 32 chunks of K=4 -> K=128
      v2f a = *(const v2f*)(rowp + kc * 4);    // global_load_b64 (8B aligned)
      v2f b;                                   // broadcast anchor chunk
      b.x = sA[kc * 4 + half * 2 + 0];
      b.y = sA[kc * 4 + half * 2 + 1];
      c = __builtin_amdgcn_wmma_f32_16x16x4_f32(
          /*neg_a=*/false, a, /*neg_b=*/false, b,
          /*c_mod=*/(short)0, c, /*reuse_a=*/false, /*reuse_b=*/false);
    }
    // Every (lane, v) cell of D holds dot[row]; columns are identical, so the
    // full-wave sum counts each of the 16 rows exactly 16 times.
    float s = 0.0f;
#pragma unroll
    for (int v = 0; v < 8; ++v) {
      float d = ps - c[v] + MARGIN;
      s += (d > 0.0f) ? d : 0.0f;
    }
    acc += s;
  }

#pragma unroll
  for (int m = 16; m >= 1; m >>= 1) acc += __shfl_xor(acc, m, 32);
  if (lane == 0) partials[gwave] = acc * (1.0f / 16.0f);
}

// Kernel 2: deterministic fixed-order reduction of partials + scalar tail
// (N not divisible by 16) + final divide by N.
__global__ __launch_bounds__(256) void triplet_finalize(
    const float* __restrict__ partials, int nparts,
    const float* __restrict__ anchor,
    const float* __restrict__ positive,
    const float* __restrict__ negatives,
    int N, int tail_start, float* __restrict__ out) {
  __shared__ float red[256];
  float s = 0.0f;
  for (int i = threadIdx.x; i < nparts; i += 256) s += partials[i];
  red[threadIdx.x] = s;
  __syncthreads();
  for (int off = 128; off >= 1; off >>= 1) {
    if (threadIdx.x < off) red[threadIdx.x] += red[threadIdx.x + off];
    __syncthreads();
  }
  if (threadIdx.x == 0) {
    float total = red[0];
    if (tail_start < N) {
      float ps = 0.0f;
      for (int k = 0; k < 128; ++k) ps += anchor[k] * positive[k];
      for (int r = tail_start; r < N; ++r) {
        float d = 0.0f;
        for (int k = 0; k < 128; ++k) d += negatives[(size_t)r * 128 + k] * anchor[k];
        float L = ps - d + MARGIN;
        total += (L > 0.0f) ? L : 0.0f;
      }
    }
    out[0] = total / (float)N;
  }
}

extern "C" void kernel_launch(void* const* d_in, const int* in_sizes, int n_in,
                              void* d_out, int out_size, void* d_ws, size_t ws_size,
                              hipStream_t stream) {
  const float* anchor = (const float*)d_in[0];     // [1,128]
  const float* positive = (const float*)d_in[1];   // [1,128]
  const float* negatives = (const float*)d_in[2];  // [N,128]
  const int D = 128;
  const int N = in_sizes[2] / D;
  const int n_tiles = N / 16;        // 31250 at N = 500000
  const int tail_start = n_tiles * 16;

  // ~1 tile per wave: 8 waves per 256-thread block.
  int blocks = (n_tiles + 7) / 8;
  if (blocks < 1) blocks = 1;
  // Fit one float partial per wave in d_ws; shrink grid if scratch is small
  // (the grid-stride loop in kernel 1 covers the remainder).
  size_t max_waves = ws_size / sizeof(float);
  if ((size_t)blocks * 8 > max_waves) {
    blocks = (int)(max_waves / 8);
    if (blocks < 1) blocks = 1;
  }
  const int nwaves = blocks * 8;
  float* partials = (float*)d_ws;

  triplet_wmma_partial<<<blocks, 256, 0, stream>>>(anchor, positive, negatives,
                                                   partials, n_tiles);
  triplet_finalize<<<1, 256, 0, stream>>>(partials, nwaves, anchor, positive,
                                          negatives, N, tail_start,
                                          (float*)d_out);
}